// CAttention_71038759075975
// MI455X (gfx1250) — compile-verified
//
#include <hip/hip_runtime.h>
#include <hip/hip_bf16.h>
#include <math.h>
#include <stdint.h>

// ---------------------------------------------------------------------------
// Channel ("transposed") attention, fp32 end-to-end on CDNA5 WMMA,
// with gfx1250 async global->LDS staging (ASYNCcnt path).
//   x:(4,384,128,128)  qkv_w:(1152,384)  proj_w:(384,384)  temperature:(8,1,1)
// ---------------------------------------------------------------------------

typedef __attribute__((ext_vector_type(2))) float v2f;
typedef __attribute__((ext_vector_type(8))) float v8f;
typedef __attribute__((ext_vector_type(4))) int   v4i;

#define BATCH 4
#define CCH   384
#define HEADS 8
#define CH    48
#define NPIX  16384          // 128*128
#define OC3   1152           // 3*C
#define EPSN  1e-12f

// ----- CDNA5 WMMA: V_WMMA_F32_16X16X4_F32 ---------------------------------
__device__ __forceinline__ v8f wmma4(v2f a, v2f b, v8f c) {
    return __builtin_amdgcn_wmma_f32_16x16x4_f32(
        /*neg_a=*/false, a, /*neg_b=*/false, b,
        /*c_mod=*/(short)0, c, /*reuse_a=*/false, /*reuse_b=*/false);
}

// ----- gfx1250 async global->LDS copy (GLOBAL_LOAD_ASYNC_TO_LDS_B128) ------
// clang-22 signature: (AS1 int4* src, AS3 int4* dst, int offset, int cpol)
__device__ __forceinline__ void async_copy_b128(const float* gptr, float* lptr) {
#if __has_builtin(__builtin_amdgcn_global_load_async_to_lds_b128)
    __builtin_amdgcn_global_load_async_to_lds_b128(
        (__attribute__((address_space(1))) v4i*)gptr,
        (__attribute__((address_space(3))) v4i*)lptr,
        /*offset=*/0, /*cpol=*/0);
#else
    unsigned lofs =
        (unsigned)(uintptr_t)(__attribute__((address_space(3))) void*)lptr;
    asm volatile("global_load_async_to_lds_b128 %0, %1, off"
                 :: "v"(lofs), "v"(gptr) : "memory");
#endif
}

__device__ __forceinline__ void wait_async0() {
#if __has_builtin(__builtin_amdgcn_s_wait_asynccnt)
    __builtin_amdgcn_s_wait_asynccnt(0);
#else
    asm volatile("s_wait_asynccnt 0x0" ::: "memory");
#endif
}

// ---------------------------------------------------------------------------
// Kernel 1/5: Out[b][m][n] = sum_k W[m][k] * X[b][k][n]
// grid: (N/128, M/64, B), block: 128 threads (4 waves); wave w owns rows
// m0+16w .. m0+16w+15 and an N-tile of 128 (8 sub-tiles of 16).
// X slab (32 x 128) staged into LDS via async global->LDS; weights from L2.
// ---------------------------------------------------------------------------
__global__ __launch_bounds__(128)
void wgemm_kernel(const float* __restrict__ W, const float* __restrict__ X,
                  float* __restrict__ Out, int M, int K, int N) {
    __shared__ __align__(16) float lx[32 * 128];   // 16 KB

    const int lane  = threadIdx.x & 31;
    const int wave  = threadIdx.x >> 5;
    const int row   = lane & 15;       // M-row (A) / N-col (B,D) within a tile
    const int khalf = lane >> 4;       // 0|1 -> K pair selector

    const int n0 = blockIdx.x * 128;
    const int m0 = blockIdx.y * 64 + wave * 16;
    const float* Xb = X + (size_t)blockIdx.z * K * N;
    float*       Ob = Out + (size_t)blockIdx.z * M * N;

    v8f acc[8];
#pragma unroll
    for (int t = 0; t < 8; ++t) acc[t] = (v8f)0.0f;

    for (int k0 = 0; k0 < K; k0 += 32) {
        __syncthreads();   // previous slab fully consumed before overwrite
        // stage X[k0..k0+32)[n0..n0+128) -> LDS: 1024 float4 / 128 threads,
        // each thread issues 8 async b128 copies (no VGPR round-trip).
#pragma unroll
        for (int i = 0; i < 8; ++i) {
            int idx = threadIdx.x + i * 128;
            int r   = idx >> 5;           // 0..31  (k row)
            int c4  = idx & 31;           // float4 column
            async_copy_b128(Xb + (size_t)(k0 + r) * N + n0 + c4 * 4,
                            lx + r * 128 + c4 * 4);
        }
        if (k0 + 32 < K)   // pull next slab toward the caches
            __builtin_prefetch(Xb + (size_t)(k0 + 32) * N + n0 + lane * 16, 0, 1);
        wait_async0();     // this wave's copies landed in LDS
        __syncthreads();   // all waves' copies visible

#pragma unroll
        for (int kk = 0; kk < 32; kk += 4) {
            const int kp = kk + 2 * khalf;
            // A fragment: W[m0+row][k0+kp .. +1]
            v2f a = *(const v2f*)(W + (size_t)(m0 + row) * K + k0 + kp);
#pragma unroll
            for (int t = 0; t < 8; ++t) {
                v2f b;
                b.x = lx[kp * 128 + t * 16 + row];
                b.y = lx[(kp + 1) * 128 + t * 16 + row];
                acc[t] = wmma4(a, b, acc[t]);
            }
        }
    }

    // D layout: lane l, vgpr v -> (M = v + 8*(l>=16), N = l%16)
#pragma unroll
    for (int t = 0; t < 8; ++t)
#pragma unroll
        for (int v = 0; v < 8; ++v)
            Ob[(size_t)(m0 + v + 8 * khalf) * N + n0 + t * 16 + row] = acc[t][v];
}

// ---------------------------------------------------------------------------
// Kernel 2: L2-norm denominators for q and k rows.
// grid.x = B*768 (first 384 rows per batch = q, next 384 = k), block = 256
// norms[b*768 + r] = max(sqrt(sum_n row^2), EPS)
// ---------------------------------------------------------------------------
__global__ __launch_bounds__(256)
void rownorm_kernel(const float* __restrict__ qkv, float* __restrict__ norms) {
    const int r   = blockIdx.x;            // 0 .. B*768-1
    const int b   = r / 768;
    const int rem = r % 768;               // 0..383 q rows, 384..767 k rows
    const float* p = qkv + ((size_t)b * OC3 + rem) * NPIX;

    float s = 0.f;
    for (int i = threadIdx.x; i < NPIX; i += 256) {
        float x = p[i];
        s += x * x;
    }
#pragma unroll
    for (int m = 16; m > 0; m >>= 1) s += __shfl_xor(s, m, 32);

    __shared__ float red[8];
    if ((threadIdx.x & 31) == 0) red[threadIdx.x >> 5] = s;
    __syncthreads();
    if (threadIdx.x == 0) {
        float tot = 0.f;
#pragma unroll
        for (int i = 0; i < 8; ++i) tot += red[i];
        norms[r] = fmaxf(sqrtf(tot), EPSN);
    }
}

// ---------------------------------------------------------------------------
// Kernel 3: per (b,h) Gram matrix S = q k^T (48x48, K=16384) + softmax.
// grid.x = B*HEADS, block = 256 (8 waves). Each wave reduces a K-slice of
// 2048 into a 3x3 grid of 16x16 WMMA tiles; partials summed in LDS via
// ds_add_f32; then 48 threads apply norm/temperature scaling + softmax.
// ---------------------------------------------------------------------------
__global__ __launch_bounds__(256)
void gram_softmax_kernel(const float* __restrict__ qkv,
                         const float* __restrict__ norms,
                         const float* __restrict__ temperature,
                         float* __restrict__ attn) {
    const int bh = blockIdx.x;
    const int b  = bh >> 3;
    const int h  = bh & 7;
    const float* Q  = qkv + ((size_t)b * OC3 + h * CH) * NPIX;
    const float* Km = qkv + ((size_t)b * OC3 + CCH + h * CH) * NPIX;

    __shared__ float S[CH * CH];           // 9216 B
    for (int i = threadIdx.x; i < CH * CH; i += 256) S[i] = 0.f;
    __syncthreads();

    const int lane  = threadIdx.x & 31;
    const int wave  = threadIdx.x >> 5;
    const int row   = lane & 15;
    const int khalf = lane >> 4;

    v8f acc[3][3];
#pragma unroll
    for (int i = 0; i < 3; ++i)
#pragma unroll
        for (int j = 0; j < 3; ++j) acc[i][j] = (v8f)0.0f;

    const int kbase = wave * 2048;
    for (int kk = 0; kk < 2048; kk += 4) {
        const int kg = kbase + kk + 2 * khalf;
        v2f a[3], bb[3];
#pragma unroll
        for (int i = 0; i < 3; ++i)
            a[i] = *(const v2f*)(Q + (size_t)(i * 16 + row) * NPIX + kg);
#pragma unroll
        for (int j = 0; j < 3; ++j)
            bb[j] = *(const v2f*)(Km + (size_t)(j * 16 + row) * NPIX + kg);
#pragma unroll
        for (int i = 0; i < 3; ++i)
#pragma unroll
            for (int j = 0; j < 3; ++j)
                acc[i][j] = wmma4(a[i], bb[j], acc[i][j]);
    }

    // LDS float-atomic reduction of the 8 wave-partials (ds_add_f32)
#pragma unroll
    for (int i = 0; i < 3; ++i)
#pragma unroll
        for (int j = 0; j < 3; ++j)
#pragma unroll
            for (int v = 0; v < 8; ++v)
                atomicAdd(&S[(i * 16 + v + 8 * khalf) * CH + j * 16 + row],
                          acc[i][j][v]);
    __syncthreads();

    // softmax over d for each of the 48 rows
    const float temp = temperature[h];
    if (threadIdx.x < CH) {
        const int c = threadIdx.x;
        const float nq = norms[b * 768 + h * CH + c];
        float mx = -INFINITY;
        for (int d = 0; d < CH; ++d) {
            float nk = norms[b * 768 + 384 + h * CH + d];
            float lv = S[c * CH + d] * temp / (nq * nk);
            S[c * CH + d] = lv;
            mx = fmaxf(mx, lv);
        }
        float sum = 0.f;
        for (int d = 0; d < CH; ++d) {
            float e = expf(S[c * CH + d] - mx);
            S[c * CH + d] = e;
            sum += e;
        }
        const float inv = 1.f / sum;
        float* ap = attn + (size_t)bh * CH * CH + (size_t)c * CH;
        for (int d = 0; d < CH; ++d) ap[d] = S[c * CH + d] * inv;
    }
}

// ---------------------------------------------------------------------------
// Kernel 4: out1[b][h*48+c][n] = sum_d attn[bh][c][d] * v[b][h][d][n]
// grid: (N/256, B*HEADS), block 256 (8 waves); each wave: 3 M-tiles x 2
// N-tiles (48 x 32), K = 48 (12 WMMA steps), attn staged in LDS (async).
// ---------------------------------------------------------------------------
__global__ __launch_bounds__(256)
void av_kernel(const float* __restrict__ qkv, const float* __restrict__ attn,
               float* __restrict__ out1) {
    const int bh = blockIdx.y;
    const int b  = bh >> 3;
    const int h  = bh & 7;
    const float* V = qkv + ((size_t)b * OC3 + 2 * CCH + h * CH) * NPIX;
    float*       O = out1 + ((size_t)b * CCH + h * CH) * NPIX;

    __shared__ __align__(16) float sA[CH * CH];   // 48*48 = 2304 floats
    {
        const float* ap = attn + (size_t)bh * CH * CH;
        // 576 float4 chunks staged async
        for (int i = threadIdx.x; i < 576; i += 256)
            async_copy_b128(ap + i * 4, sA + i * 4);
        wait_async0();
    }
    __syncthreads();

    const int lane  = threadIdx.x & 31;
    const int wave  = threadIdx.x >> 5;
    const int row   = lane & 15;
    const int khalf = lane >> 4;
    const int n0    = blockIdx.x * 256 + wave * 32;

    v8f acc[3][2];
#pragma unroll
    for (int i = 0; i < 3; ++i)
#pragma unroll
        for (int t = 0; t < 2; ++t) acc[i][t] = (v8f)0.0f;

#pragma unroll
    for (int kk = 0; kk < CH; kk += 4) {
        const int kp = kk + 2 * khalf;
        v2f a[3];
#pragma unroll
        for (int i = 0; i < 3; ++i) {
            a[i].x = sA[(i * 16 + row) * CH + kp];
            a[i].y = sA[(i * 16 + row) * CH + kp + 1];
        }
        v2f bb[2];
#pragma unroll
        for (int t = 0; t < 2; ++t) {
            bb[t].x = V[(size_t)kp * NPIX + n0 + t * 16 + row];
            bb[t].y = V[(size_t)(kp + 1) * NPIX + n0 + t * 16 + row];
        }
#pragma unroll
        for (int i = 0; i < 3; ++i)
#pragma unroll
            for (int t = 0; t < 2; ++t)
                acc[i][t] = wmma4(a[i], bb[t], acc[i][t]);
    }

#pragma unroll
    for (int i = 0; i < 3; ++i)
#pragma unroll
        for (int t = 0; t < 2; ++t)
#pragma unroll
            for (int v = 0; v < 8; ++v)
                O[(size_t)(i * 16 + v + 8 * khalf) * NPIX + n0 + t * 16 + row]
                    = acc[i][t][v];
}

// ---------------------------------------------------------------------------
// Host-side launch
// ---------------------------------------------------------------------------
extern "C" void kernel_launch(void* const* d_in, const int* in_sizes, int n_in,
                              void* d_out, int out_size, void* d_ws, size_t ws_size,
                              hipStream_t stream) {
    const float* x      = (const float*)d_in[0];   // (4,384,128,128)
    const float* qkv_w  = (const float*)d_in[1];   // (1152,384)
    const float* proj_w = (const float*)d_in[2];   // (384,384)
    const float* temp   = (const float*)d_in[3];   // (8,1,1)
    float*       out    = (float*)d_out;           // (4,384,128,128)

    // workspace layout (floats)
    float* ws    = (float*)d_ws;
    float* qkv   = ws;                                           // B*1152*N
    float* out1  = qkv + (size_t)BATCH * OC3 * NPIX;             // B*384*N
    float* norms = out1 + (size_t)BATCH * CCH * NPIX;            // B*768
    float* attn  = norms + (size_t)BATCH * 768;                  // B*8*48*48

    // 1) qkv = qkv_w @ x   (per batch)
    {
        dim3 grid(NPIX / 128, OC3 / 64, BATCH), block(128);
        wgemm_kernel<<<grid, block, 0, stream>>>(qkv_w, x, qkv, OC3, CCH, NPIX);
    }
    // 2) q,k row norms
    {
        dim3 grid(BATCH * 768), block(256);
        rownorm_kernel<<<grid, block, 0, stream>>>(qkv, norms);
    }
    // 3) Gram + softmax -> attn
    {
        dim3 grid(BATCH * HEADS), block(256);
        gram_softmax_kernel<<<grid, block, 0, stream>>>(qkv, norms, temp, attn);
    }
    // 4) out1 = attn @ v
    {
        dim3 grid(NPIX / 256, BATCH * HEADS), block(256);
        av_kernel<<<grid, block, 0, stream>>>(qkv, attn, out1);
    }
    // 5) out = proj_w @ out1  (per batch)
    {
        dim3 grid(NPIX / 128, CCH / 64, BATCH), block(128);
        wgemm_kernel<<<grid, block, 0, stream>>>(proj_w, out1, out, CCH, CCH, NPIX);
    }
}